// OpticalConv2dNew_69698729279498
// MI455X (gfx1250) — compile-verified
//
#include <hip/hip_runtime.h>
#include <stdint.h>

typedef __attribute__((ext_vector_type(2))) float v2f;
typedef __attribute__((ext_vector_type(4))) float v4f;
typedef __attribute__((ext_vector_type(8))) float v8f;

#define HH 128
#define WW 128
#define NB 8
#define NC 16
#define NO 32
#define KS 9
#define NSTEP 27          // K = 9 rows * 12 padded cols = 108 taps = 27 WMMA steps (K=4)
#define NPAIR 54          // 108/2 tap-pairs
#define XS 24             // 16 pixel rows + 8 circular halo rows
#define XCOLS 27          // 16 + 8 halo + 3 pad cols (pad holds real wrapped data)
#define XST 28            // LDS row stride in floats
#define XELEMS (XS * XCOLS)          // 648 staged elements per channel
#define XBUFB  (XS * XST * 4)        // bytes per x buffer
#define CHB    (HH * WW * 4)         // bytes per input channel image

__global__ __launch_bounds__(512) void optical_conv_wmma(
    const float* __restrict__ x,     // [8,16,128,128]
    const float* __restrict__ kern,  // [32,16,9,9] (only in-channel 0 used, per reference)
    float* __restrict__ out)         // [8,32,128,128]
{
  // A fragments packed for one ds_load_b128 per step:
  // kerAB[(2s+half)*16 + n] = { w_t(o=n), w_t+1(o=n), w_t(o=n+16), w_t+1(o=n+16) }
  __shared__ __align__(16) float kerAB[NPAIR * 16 * 4];   // 13.5 KB
  __shared__ float xsbuf[2][XS * XST];                    // 5.25 KB, double-buffered

  const int tid  = threadIdx.x;
  const int wave = tid >> 5;               // pixel row r within the 16x16 tile
  const int lane = tid & 31;
  const int n    = lane & 15;              // pixel col (B/D N) and o-row (A M)
  const int half = lane >> 4;              // K-slot selector within a step

  const int b   = blockIdx.z;
  const int uu0 = blockIdx.y * 16;
  const int vv0 = blockIdx.x * 16;

  // ---- stage reordered, row-padded, tile-interleaved kernel weights ----
  for (int j = tid; j < NPAIR * 16; j += 512) {
    int p = j >> 4, n16 = j & 15;
    int t = 2 * p;                          // even tap; pair stays inside one 12-col row
    int arow = t / 12, bcol = t % 12;
    const float* kr0 = kern + (size_t)n16 * (NC * KS * KS) + arow * KS;
    const float* kr1 = kern + (size_t)(n16 + 16) * (NC * KS * KS) + arow * KS;
    kerAB[j * 4 + 0] = (bcol     < KS) ? kr0[bcol]     : 0.0f;
    kerAB[j * 4 + 1] = (bcol + 1 < KS) ? kr0[bcol + 1] : 0.0f;
    kerAB[j * 4 + 2] = (bcol     < KS) ? kr1[bcol]     : 0.0f;
    kerAB[j * 4 + 3] = (bcol + 1 < KS) ? kr1[bcol + 1] : 0.0f;
  }

  // ---- per-thread async-staging slots (each thread owns <=2 elements) ----
  const unsigned ldsbase = (unsigned)(uintptr_t)&xsbuf[0][0];
  const float* xsrc0 = x + (size_t)b * NC * HH * WW;     // channel 0 of this batch

  const int i0 = tid;                       // always < 648
  const int ly0 = i0 / XCOLS, lx0 = i0 % XCOLS;
  const unsigned lds0 = ldsbase + (unsigned)(ly0 * XST + lx0) * 4;
  uint64_t g0 = (uint64_t)(uintptr_t)(xsrc0 +
      (((uu0 - 3 + ly0) & (HH - 1)) * WW) + ((vv0 - 6 + lx0) & (WW - 1)));

  const int i1 = tid + 512;
  const bool s1ok = (i1 < XELEMS);
  const int ly1 = i1 / XCOLS, lx1 = i1 % XCOLS;
  const unsigned lds1 = ldsbase + (unsigned)(ly1 * XST + lx1) * 4;
  uint64_t g1 = (uint64_t)(uintptr_t)(xsrc0 +
      (((uu0 - 3 + ly1) & (HH - 1)) * WW) + ((vv0 - 6 + lx1) & (WW - 1)));

  // ---- prologue: async-fill buffer 0 with channel 0 ----
  asm volatile("global_load_async_to_lds_b32 %0, %1, off"
               :: "v"(lds0), "v"(g0) : "memory");
  if (s1ok)
    asm volatile("global_load_async_to_lds_b32 %0, %1, off"
                 :: "v"(lds1), "v"(g1) : "memory");
  g0 += CHB; g1 += CHB;

  // per-lane compute bases (all inner-loop LDS accesses = base + const immediate)
  const v4f* ka = (const v4f*)kerAB + half * 16 + n;     // + s*32 per step
  const int xoff_lane = wave * XST + (n + 3 - 2 * half);

  v8f acc0 = {0,0,0,0,0,0,0,0};            // sum_c |field|, o = 0..15
  v8f acc1 = {0,0,0,0,0,0,0,0};            // o = 16..31

  for (int c = 0; c < NC; ++c) {
    asm volatile("s_wait_asynccnt 0x0" ::: "memory");   // this wave's fills done
    __syncthreads();                                    // everyone's fills done

    if (c + 1 < NC) {                      // overlap: fill next buffer during compute
      const unsigned boff = ((c + 1) & 1) ? (unsigned)XBUFB : 0u;
      asm volatile("global_load_async_to_lds_b32 %0, %1, off"
                   :: "v"(lds0 + boff), "v"(g0) : "memory");
      if (s1ok)
        asm volatile("global_load_async_to_lds_b32 %0, %1, off"
                     :: "v"(lds1 + boff), "v"(g1) : "memory");
      g0 += CHB; g1 += CHB;
    }

    const float* xb = xsbuf[c & 1] + xoff_lane;

    v8f c0 = {0,0,0,0,0,0,0,0};
    v8f c1 = {0,0,0,0,0,0,0,0};

    #pragma unroll
    for (int s = 0; s < NSTEP; ++s) {
      const int arow = s / 3;              // compile-time under full unroll
      const int cg   = (s % 3) * 4;
      const int off  = (8 - arow) * XST + (8 - cg);
      v2f bf;
      bf.x = xb[off];                      // tap col cg + 2*half
      bf.y = xb[off - 1];                  // tap col cg + 2*half + 1 (one col left)
      v4f av = ka[s * 32];                 // one ds_load_b128: both o-tiles' A pairs
      v2f a0; a0.x = av.x; a0.y = av.y;
      v2f a1; a1.x = av.z; a1.y = av.w;
      c0 = __builtin_amdgcn_wmma_f32_16x16x4_f32(false, a0, false, bf,
                                                 (short)0, c0, false, false);
      c1 = __builtin_amdgcn_wmma_f32_16x16x4_f32(false, a1, false, bf,
                                                 (short)0, c1, false, false);
    }

    #pragma unroll
    for (int j = 0; j < 8; ++j) {
      acc0[j] += __builtin_fabsf(c0[j]);
      acc1[j] += __builtin_fabsf(c1[j]);
    }
  }

  // D layout: VGPR j <-> o = j + 8*half (tile-local), lane n <-> pixel col
  float* outp = out + ((size_t)(b * NO) * HH + (uu0 + wave)) * WW + (vv0 + n);
  #pragma unroll
  for (int j = 0; j < 8; ++j) {
    int o0 = j + 8 * half;
    outp[(size_t)o0 * (HH * WW)]        = acc0[j];
    outp[(size_t)(16 + o0) * (HH * WW)] = acc1[j];
  }
}

extern "C" void kernel_launch(void* const* d_in, const int* in_sizes, int n_in,
                              void* d_out, int out_size, void* d_ws, size_t ws_size,
                              hipStream_t stream) {
  const float* x    = (const float*)d_in[0];
  const float* kern = (const float*)d_in[1];
  float* out        = (float*)d_out;
  dim3 grid(WW / 16, HH / 16, NB);         // 512 workgroups
  dim3 block(512);                         // 16 wave32 waves
  hipLaunchKernelGGL(optical_conv_wmma, grid, block, 0, stream, x, kern, out);
}